// mdcm_eqx_87875030876986
// MI455X (gfx1250) — compile-verified
//
#include <hip/hip_runtime.h>

#define CLIPV 0.173f
#define AS1 __attribute__((address_space(1)))
#define AS3 __attribute__((address_space(3)))

typedef __attribute__((ext_vector_type(2))) float v2f;
typedef __attribute__((ext_vector_type(8))) float v8f;
typedef int v4i_b128 __attribute__((vector_size(4 * sizeof(int))));

// ---------------------------------------------------------------------------
// Exact 32-lane sum using V_WMMA_F32_16X16X4_F32.
// A (16x4): lane m supplies A[m][0] (V0) / A[m][1] (V1); lane m+16 supplies
// A[m][2]/A[m][3].  Put acc in V0, 0 in V1; B = all-ones (4x16).
// D[m][n] = sum_k A[m][k] = acc[m] + acc[m+16]  (independent of n).
// Summing the 8 D VGPRs per lane + one shfl_xor(16) completes the reduction.
// Must be executed with full EXEC (all 32 lanes) - called from uniform code.
// ---------------------------------------------------------------------------
__device__ __forceinline__ float wave_sum_wmma(float acc) {
    v2f a; a[0] = acc;  a[1] = 0.0f;
    v2f b; b[0] = 1.0f; b[1] = 1.0f;
    v8f c = {};
    v8f d = __builtin_amdgcn_wmma_f32_16x16x4_f32(
        /*neg_a=*/false, a, /*neg_b=*/false, b,
        /*c_mod=*/(short)0, c, /*reuse_a=*/false, /*reuse_b=*/false);
    float t = ((d[0] + d[1]) + (d[2] + d[3])) + ((d[4] + d[5]) + (d[6] + d[7]));
    t += __shfl_xor(t, 16, 32);
    return t; // every lane holds the 32-lane sum
}

// ---------------------------------------------------------------------------
// K1: seg[s] = sum_{i: chg_idx[i]==s} x0[chg_typ_idx[i]] * charges[i], s in {0,1}
// Single block, deterministic LDS tree reduction.
// ---------------------------------------------------------------------------
__global__ void k_segsum(const float* __restrict__ x0,
                         const float* __restrict__ charges,
                         const int*   __restrict__ chg_idx,
                         const int*   __restrict__ chg_typ_idx,
                         int nchg, float* __restrict__ seg_out) {
    __shared__ float s0[256];
    __shared__ float s1[256];
    float a0 = 0.0f, a1 = 0.0f;
    for (int i = threadIdx.x; i < nchg; i += blockDim.x) {
        float v = x0[chg_typ_idx[i]] * charges[i];
        if (chg_idx[i] == 0) a0 += v; else a1 += v;
    }
    s0[threadIdx.x] = a0; s1[threadIdx.x] = a1;
    __syncthreads();
    for (int off = 128; off > 0; off >>= 1) {
        if ((int)threadIdx.x < off) {
            s0[threadIdx.x] += s0[threadIdx.x + off];
            s1[threadIdx.x] += s1[threadIdx.x + off];
        }
        __syncthreads();
    }
    if (threadIdx.x == 0) { seg_out[0] = s0[0]; seg_out[1] = s1[0]; }
}

// ---------------------------------------------------------------------------
// K2: per-charge global position + constrained charge -> packed float4 table.
// charge i: frame f=i/6, atom a=(i>>1)%3, sub-charge c=i&1.
// ---------------------------------------------------------------------------
__global__ void k_positions(const float* __restrict__ x0,
                            const float* __restrict__ coords,
                            const float* __restrict__ charges,
                            const int*   __restrict__ chg_idx,
                            const int*   __restrict__ chg_typ_idx,
                            const int*   __restrict__ local_typ_idx,
                            const int*   __restrict__ p_nchgparm,
                            const int*   __restrict__ p_ncharges,
                            const float* __restrict__ seg,
                            int nchg, float4* __restrict__ table) {
    int i = blockIdx.x * blockDim.x + threadIdx.x;
    if (i >= nchg) return;
    int   nchgparm = p_nchgparm[0];
    float inv_nc   = 1.0f / (float)p_ncharges[0];

    float q = (x0[chg_typ_idx[i]] - seg[chg_idx[i]] * inv_nc) * charges[i];

    int f = i / 6;
    int a = (i >> 1) % 3;
    const float* fr = coords + f * 9;
    int an = (a + 1) % 3, ap = (a + 2) % 3;

    float r0 = fr[a*3+0], r1 = fr[a*3+1], r2 = fr[a*3+2];
    float ex0 = fr[an*3+0]-r0, ex1 = fr[an*3+1]-r1, ex2 = fr[an*3+2]-r2;
    float inv = rsqrtf(ex0*ex0 + ex1*ex1 + ex2*ex2);
    ex0 *= inv; ex1 *= inv; ex2 *= inv;

    float v0 = fr[ap*3+0]-r0, v1 = fr[ap*3+1]-r1, v2 = fr[ap*3+2]-r2;
    float cz0 = ex1*v2 - ex2*v1;
    float cz1 = ex2*v0 - ex0*v2;
    float cz2 = ex0*v1 - ex1*v0;
    inv = rsqrtf(cz0*cz0 + cz1*cz1 + cz2*cz2);
    float ez0 = cz0*inv, ez1 = cz1*inv, ez2 = cz2*inv;

    float ey0 = ez1*ex2 - ez2*ex1;
    float ey1 = ez2*ex0 - ez0*ex2;
    float ey2 = ez0*ex1 - ez1*ex0;

    // x0_local[i][j] = clip(x0[nchgparm + local_typ_idx[i*3+j]], -CLIP, CLIP)
    float l0 = fminf(fmaxf(x0[nchgparm + local_typ_idx[i*3+0]], -CLIPV), CLIPV);
    float l1 = fminf(fmaxf(x0[nchgparm + local_typ_idx[i*3+1]], -CLIPV), CLIPV);
    float l2 = fminf(fmaxf(x0[nchgparm + local_typ_idx[i*3+2]], -CLIPV), CLIPV);

    // disp_i = R[i][:] . [l0,l1,l2] with columns (ex, ey, ez)
    float px = r0 + ex0*l0 + ey0*l1 + ez0*l2;
    float py = r1 + ex1*l0 + ey1*l1 + ez1*l2;
    float pz = r2 + ex2*l0 + ey2*l1 + ez2*l2;
    table[i] = make_float4(px, py, pz, q);
}

// ---------------------------------------------------------------------------
// K3: main streaming kernel. Stage the entire 192KB charge table in LDS
// (CDNA5: 320KB per WGP) using GLOBAL_LOAD_ASYNC_TO_LDS_B128 (ASYNCcnt path,
// no VGPR round-trip), then stream a contiguous chunk of grid points.
// Deterministic partial per block; WMMA wave reduction.
// ---------------------------------------------------------------------------
__global__ void __launch_bounds__(1024, 1)
k_esp(const float4* __restrict__ table, int table_n,
      const float* __restrict__ grids,
      const float* __restrict__ esp_ref,
      const int*   __restrict__ grid_idx,
      int G, int nblocks,
      float* __restrict__ partials) {
    extern __shared__ float4 lds[];            // table_n float4 + 32 floats
    float* wpart = (float*)(lds + table_n);

    // async memory -> LDS fill (per-lane LDS dest address, 16B each)
    for (int i = threadIdx.x; i < table_n; i += blockDim.x) {
        __builtin_amdgcn_global_load_async_to_lds_b128(
            (AS1 v4i_b128*)(table + i),
            (AS3 v4i_b128*)(lds + i),
            /*offset=*/0, /*cpol=*/0);
    }
    asm volatile("s_wait_asynccnt 0x0" ::: "memory");
    __syncthreads();

    int chunk = (G + nblocks - 1) / nblocks;
    int start = blockIdx.x * chunk;
    int end   = start + chunk; if (end > G) end = G;

    const float3* __restrict__ g3 = (const float3*)grids;

    float acc = 0.0f;
    for (int g = start + (int)threadIdx.x; g < end; g += blockDim.x) {
        int gp = g + (int)blockDim.x; if (gp >= end) gp = g;   // clamped prefetch addr
        __builtin_prefetch(&g3[gp], 0, 1);
        __builtin_prefetch(&grid_idx[gp], 0, 1);

        int    f   = grid_idx[g];
        float3 gv  = g3[g];
        float  ref = esp_ref[g];
        const float4* __restrict__ p = lds + f * 6;
        float e = 0.0f;
#pragma unroll
        for (int c = 0; c < 6; ++c) {
            float4 pc = p[c];
            float dx = gv.x - pc.x, dy = gv.y - pc.y, dz = gv.z - pc.z;
            e += pc.w * rsqrtf(dx*dx + dy*dy + dz*dz);
        }
        float diff = e - ref;
        acc += diff * diff;
    }

    // full-EXEC wave reduction (uniform code path, block is multiple of 32)
    float t = wave_sum_wmma(acc);
    if ((threadIdx.x & 31) == 0) wpart[threadIdx.x >> 5] = t;
    __syncthreads();
    if (threadIdx.x == 0) {
        float s = 0.0f;
        int nw = blockDim.x >> 5;
        for (int w = 0; w < nw; ++w) s += wpart[w];
        partials[blockIdx.x] = s;
    }
}

// ---------------------------------------------------------------------------
// K4: final deterministic reduction of block partials, scale, write output.
// ---------------------------------------------------------------------------
__global__ void k_final(const float* __restrict__ partials, int n,
                        float* __restrict__ out) {
    __shared__ float s[256];
    float a = 0.0f;
    for (int i = threadIdx.x; i < n; i += 256) a += partials[i];
    s[threadIdx.x] = a;
    __syncthreads();
    for (int off = 128; off > 0; off >>= 1) {
        if ((int)threadIdx.x < off) s[threadIdx.x] += s[threadIdx.x + off];
        __syncthreads();
    }
    if (threadIdx.x == 0) out[0] = s[0] * 1.0e-6f;
}

// ---------------------------------------------------------------------------
extern "C" void kernel_launch(void* const* d_in, const int* in_sizes, int n_in,
                              void* d_out, int out_size, void* d_ws, size_t ws_size,
                              hipStream_t stream) {
    const float* x0            = (const float*)d_in[0];
    const float* all_coords    = (const float*)d_in[1];
    const float* all_grids     = (const float*)d_in[2];
    const float* all_esp       = (const float*)d_in[3];
    const float* charges       = (const float*)d_in[4];
    // d_in[5] = all_atoms (unused by the reference math)
    const int*   chg_idx       = (const int*)d_in[6];
    const int*   grid_idx      = (const int*)d_in[7];
    const int*   chg_typ_idx   = (const int*)d_in[8];
    const int*   local_typ_idx = (const int*)d_in[9];
    const int*   p_nchgparm    = (const int*)d_in[10];
    const int*   p_ncharges    = (const int*)d_in[12];

    const int NCHG = in_sizes[4];   // 12000 = F*6
    const int G    = in_sizes[3];   // 4,000,000

    // workspace layout
    float*  wsF      = (float*)d_ws;
    float*  seg      = wsF;                                  // 2 floats
    const int NB     = 512;
    float*  partials = wsF + 8;                              // NB floats
    float4* table    = (float4*)((char*)d_ws + 4096);        // NCHG float4s

    k_segsum<<<1, 256, 0, stream>>>(x0, charges, chg_idx, chg_typ_idx, NCHG, seg);

    k_positions<<<(NCHG + 255) / 256, 256, 0, stream>>>(
        x0, all_coords, charges, chg_idx, chg_typ_idx, local_typ_idx,
        p_nchgparm, p_ncharges, seg, NCHG, table);

    size_t ldsBytes = (size_t)NCHG * sizeof(float4) + 32 * sizeof(float);
    k_esp<<<NB, 1024, ldsBytes, stream>>>(
        table, NCHG, all_grids, all_esp, grid_idx, G, NB, partials);

    k_final<<<1, 256, 0, stream>>>(partials, NB, (float*)d_out);
}